// GCN_11390253269678
// MI455X (gfx1250) — compile-verified
//
#include <hip/hip_runtime.h>
#include <hip/hip_bf16.h>

typedef float v2f __attribute__((ext_vector_type(2)));
typedef float v8f __attribute__((ext_vector_type(8)));

typedef __attribute__((address_space(1))) int gas_int;   // global
typedef __attribute__((address_space(3))) int las_int;   // LDS

#define K_DIM 128

#if defined(__AMDGCN__) && \
    __has_builtin(__builtin_amdgcn_global_load_async_to_lds_b32) && \
    __has_builtin(__builtin_amdgcn_s_wait_asynccnt)
#define GCN_ASYNC_LDS 1
#endif

// ---------------------------------------------------------------------------
// GEMM: H[M x N] = X[M x 128] * W[128 x N], fp32 via V_WMMA_F32_16X16X4_F32.
// N is a template parameter so the B-fragment LDS loads fold into ds_load
// immediate offsets (kk*N*4 <= 63488 < 64K). W staged in LDS via the gfx1250
// GLOBAL_LOAD_ASYNC_TO_LDS path (ASYNCcnt + s_wait_asynccnt).
// ---------------------------------------------------------------------------
template <int N>
__global__ void __launch_bounds__(256)
gemm_wmma_f32_k(const float* __restrict__ X, const float* __restrict__ W,
                float* __restrict__ H, int M) {
    __shared__ float sW[K_DIM * N];            // 64 KB (N=128) / 32 KB (N=64)
    const int tid = threadIdx.x;
    constexpr int totW = K_DIM * N;
#if defined(GCN_ASYNC_LDS)
    for (int i = tid; i < totW; i += 256) {
        __builtin_amdgcn_global_load_async_to_lds_b32(
            (gas_int*)(int*)(W + i),
            (las_int*)(int*)(sW + i),
            0, 0);
    }
    __builtin_amdgcn_s_wait_asynccnt(0);
#else
    for (int i = tid; i < totW; i += 256) sW[i] = W[i];
#endif
    __syncthreads();

    const int lane = tid & 31;
    const int wave = tid >> 5;
    constexpr int NT = N >> 4;                 // tiles along N (8 or 4)
    const int mtiles = M >> 4;                 // 3125 (exact for M=50000)
    int t  = blockIdx.x * 8 + wave;
    int mt = t / NT;                           // constant divisor -> shifts
    int nt = t - mt * NT;
    if (mt >= mtiles) return;                  // wave-uniform: EXEC stays full

    const int half = lane >> 4;                // 0: lanes 0-15, 1: lanes 16-31
    const int l    = lane & 15;
    const int m0   = mt << 4;
    const int n0   = nt << 4;

    // A frag (16x4): lane l -> row m0+l; VGPR0 = K {kk | kk+2}, VGPR1 = K+1
    const float* Arow = X + (size_t)(m0 + l) * K_DIM + 2 * half;
    // B frag (4x16): lane -> N, (vgpr, lane-half) -> K, mirrored layout
    const float* Bp = sW + (2 * half) * N + n0 + l;
    v8f c = {};

    #pragma unroll
    for (int kk = 0; kk < K_DIM; kk += 4) {
        v2f a = *(const v2f*)(Arow + kk);
        v2f b;
        b.x = Bp[kk * N];                      // compile-time immediate offsets
        b.y = Bp[kk * N + N];
        c = __builtin_amdgcn_wmma_f32_16x16x4_f32(false, a, false, b,
                                                  (short)0, c, false, false);
    }

    // C/D layout: VGPR r -> M = r (lanes 0-15) / r+8 (lanes 16-31), N = lane&15
    float* Hp = H + (size_t)m0 * N + n0 + l;
    #pragma unroll
    for (int r = 0; r < 8; ++r)
        Hp[(size_t)(r + 8 * half) * N] = c[r];
}

// ---------------------------------------------------------------------------
// Degree: dis[i] = rsqrt(1 + #edges with dst==i)   (self-loop folded in)
// ---------------------------------------------------------------------------
__global__ void deg_init_k(float* deg, int n) {
    int i = blockIdx.x * blockDim.x + threadIdx.x;
    if (i < n) deg[i] = 1.0f;                  // self-loop contributes 1
}
__global__ void deg_accum_k(const int* __restrict__ dst, float* deg, int nE) {
    int i = blockIdx.x * blockDim.x + threadIdx.x;
    if (i < nE)
        __hip_atomic_fetch_add(&deg[dst[i]], 1.0f,
                               __ATOMIC_RELAXED, __HIP_MEMORY_SCOPE_AGENT);
}
__global__ void deg_fin_k(float* deg, int n) {
    int i = blockIdx.x * blockDim.x + threadIdx.x;
    if (i < n) deg[i] = rsqrtf(deg[i]);        // deg >= 1 always
}

__global__ void zero_k(float* p, long long n) {
    long long i = (long long)blockIdx.x * blockDim.x + threadIdx.x;
    if (i < n) p[i] = 0.0f;
}

// ---------------------------------------------------------------------------
// Edge scatter: agg[dst] += H[src] * (dis[src]*dis[dst]); float4 per thread.
// ---------------------------------------------------------------------------
__global__ void scatter_k(const float* __restrict__ H, const int* __restrict__ src,
                          const int* __restrict__ dst, const float* __restrict__ dis,
                          float* __restrict__ agg, int nE, int N) {
    const int chunks = N >> 2;
    long long idx = (long long)blockIdx.x * blockDim.x + threadIdx.x;
    long long tot = (long long)nE * chunks;
    if (idx >= tot) return;
    int e = (int)(idx / chunks);
    int c = (int)(idx - (long long)e * chunks);
    int s = src[e], d = dst[e];
    float nrm = dis[s] * dis[d];
    const float4 v = *(const float4*)(H + (size_t)s * N + 4 * c);
    float* ap = agg + (size_t)d * N + 4 * c;
    __hip_atomic_fetch_add(ap + 0, v.x * nrm, __ATOMIC_RELAXED, __HIP_MEMORY_SCOPE_AGENT);
    __hip_atomic_fetch_add(ap + 1, v.y * nrm, __ATOMIC_RELAXED, __HIP_MEMORY_SCOPE_AGENT);
    __hip_atomic_fetch_add(ap + 2, v.z * nrm, __ATOMIC_RELAXED, __HIP_MEMORY_SCOPE_AGENT);
    __hip_atomic_fetch_add(ap + 3, v.w * nrm, __ATOMIC_RELAXED, __HIP_MEMORY_SCOPE_AGENT);
}

// ---------------------------------------------------------------------------
// Epilogue: B = act( B(agg) + H*dis^2 (self-loop) + bias ), in place.
// ---------------------------------------------------------------------------
__global__ void epilogue_k(const float* __restrict__ H, const float* __restrict__ dis,
                           const float* __restrict__ bias, float* __restrict__ B,
                           int M, int N, int do_relu) {
    long long idx = (long long)blockIdx.x * blockDim.x + threadIdx.x;
    long long tot = (long long)M * N;
    if (idx >= tot) return;
    int i = (int)(idx / N);
    int j = (int)(idx - (long long)i * N);
    float di = dis[i];
    float v = B[idx] + H[idx] * di * di + bias[j];
    if (do_relu) v = fmaxf(v, 0.0f);
    B[idx] = v;
}

// ---------------------------------------------------------------------------
// Row L2-normalize (N=64): one wave32 per row, shfl_xor reduction.
// ---------------------------------------------------------------------------
__global__ void rownorm_k(const float* __restrict__ src, float* __restrict__ out, int M) {
    int wave = threadIdx.x >> 5;
    int lane = threadIdx.x & 31;
    int row = blockIdx.x * 8 + wave;
    if (row >= M) return;
    const float* p = src + (size_t)row * 64;
    float v0 = p[lane], v1 = p[lane + 32];
    float s = v0 * v0 + v1 * v1;
    #pragma unroll
    for (int off = 16; off > 0; off >>= 1) s += __shfl_xor(s, off, 32);
    float scale = 1.0f / fmaxf(sqrtf(s), 1e-12f);
    out[(size_t)row * 64 + lane]      = v0 * scale;
    out[(size_t)row * 64 + lane + 32] = v1 * scale;
}

// ---------------------------------------------------------------------------
// y branch: batch stats -> BN -> Linear(1,128)+ReLU -> Linear(128,64) -> norm
// ---------------------------------------------------------------------------
__global__ void ystats_k(const float* __restrict__ y, float* stats, int n) {
    __shared__ float ss[256], sq[256];
    int i = threadIdx.x;
    float v = (i < n) ? y[i] : 0.0f;
    ss[i] = v; sq[i] = v * v;
    __syncthreads();
    for (int off = 128; off > 0; off >>= 1) {
        if (i < off) { ss[i] += ss[i + off]; sq[i] += sq[i + off]; }
        __syncthreads();
    }
    if (i == 0) {
        float mu  = ss[0] / n;
        float var = sq[0] / n - mu * mu;       // population variance
        stats[0] = mu;
        stats[1] = rsqrtf(var + 1e-5f);
    }
}

__global__ void ymlp1_k(const float* __restrict__ y, const float* __restrict__ stats,
                        const float* __restrict__ g, const float* __restrict__ b,
                        const float* __restrict__ Wm1, const float* __restrict__ bm1,
                        float* __restrict__ T1) {
    int idx = blockIdx.x * blockDim.x + threadIdx.x;
    if (idx >= 256 * 128) return;
    int i = idx >> 7, j = idx & 127;
    float yb = (y[i] - stats[0]) * stats[1] * g[0] + b[0];
    float t = yb * Wm1[j] + bm1[j];
    T1[idx] = t > 0.0f ? t : 0.0f;
}

__global__ void __launch_bounds__(64)
ymlp2_k(const float* __restrict__ T1, const float* __restrict__ Wm2,
        const float* __restrict__ bm2, float* __restrict__ out) {
    __shared__ float sT[128];
    __shared__ float sv[64];
    int row = blockIdx.x, j = threadIdx.x;
    sT[j]      = T1[row * 128 + j];
    sT[j + 64] = T1[row * 128 + 64 + j];
    __syncthreads();
    float acc = bm2[j];
    #pragma unroll 4
    for (int k = 0; k < 128; ++k) acc += sT[k] * Wm2[k * 64 + j];
    sv[j] = acc * acc;
    __syncthreads();
    for (int off = 32; off > 0; off >>= 1) {
        if (j < off) sv[j] += sv[j + off];
        __syncthreads();
    }
    float scale = 1.0f / fmaxf(sqrtf(sv[0]), 1e-12f);
    out[(size_t)row * 64 + j] = acc * scale;
}

// ---------------------------------------------------------------------------
extern "C" void kernel_launch(void* const* d_in, const int* in_sizes, int n_in,
                              void* d_out, int out_size, void* d_ws, size_t ws_size,
                              hipStream_t stream) {
    const float* x   = (const float*)d_in[0];
    const float* y   = (const float*)d_in[1];
    const int*   ei  = (const int*)d_in[2];   // jax default: int32 (x64 off)
    const float* W1  = (const float*)d_in[3];
    const float* b1  = (const float*)d_in[4];
    const float* W2  = (const float*)d_in[5];
    const float* b2  = (const float*)d_in[6];
    const float* W3  = (const float*)d_in[7];
    const float* b3  = (const float*)d_in[8];
    const float* bng = (const float*)d_in[9];
    const float* bnb = (const float*)d_in[10];
    const float* Wm1 = (const float*)d_in[11];
    const float* bm1 = (const float*)d_in[12];
    const float* Wm2 = (const float*)d_in[13];
    const float* bm2 = (const float*)d_in[14];

    const int n = in_sizes[0] / 128;          // 50000
    const int E = in_sizes[2] / 2;            // 800000
    const int* src = ei;
    const int* dst = ei + E;

    float* ws    = (float*)d_ws;
    float* dis   = ws;                                    // n floats
    float* A     = ws + (size_t)((n + 63) & ~63);         // n*128
    float* B     = A + (size_t)n * 128;                   // n*128
    float* T1    = B + (size_t)n * 128;                   // 256*128
    float* stats = T1 + 256 * 128;                        // 2

    float* xout = (float*)d_out;
    float* yout = xout + (size_t)n * 64;

    dim3 blk(256);

    // --- degree normalization ---
    deg_init_k <<<dim3((n + 255) / 256), blk, 0, stream>>>(dis, n);
    deg_accum_k<<<dim3((E + 255) / 256), blk, 0, stream>>>(dst, dis, E);
    deg_fin_k  <<<dim3((n + 255) / 256), blk, 0, stream>>>(dis, n);

    const int mtiles = n / 16;

    // --- layer 1: x -> A -> scatter into B (N=128, ReLU) ---
    {
        const int N = 128;
        const int tiles = mtiles * (N / 16);
        gemm_wmma_f32_k<128><<<dim3((tiles + 7) / 8), blk, 0, stream>>>(x, W1, A, n);
        long long tot = (long long)n * N;
        zero_k<<<dim3((unsigned)((tot + 255) / 256)), blk, 0, stream>>>(B, tot);
        long long st = (long long)E * (N / 4);
        scatter_k<<<dim3((unsigned)((st + 255) / 256)), blk, 0, stream>>>(A, src, dst, dis, B, E, N);
        epilogue_k<<<dim3((unsigned)((tot + 255) / 256)), blk, 0, stream>>>(A, dis, b1, B, n, N, 1);
    }
    // --- layer 2: B -> A -> scatter into B (N=128, ReLU) ---
    {
        const int N = 128;
        const int tiles = mtiles * (N / 16);
        gemm_wmma_f32_k<128><<<dim3((tiles + 7) / 8), blk, 0, stream>>>(B, W2, A, n);
        long long tot = (long long)n * N;
        zero_k<<<dim3((unsigned)((tot + 255) / 256)), blk, 0, stream>>>(B, tot);
        long long st = (long long)E * (N / 4);
        scatter_k<<<dim3((unsigned)((st + 255) / 256)), blk, 0, stream>>>(A, src, dst, dis, B, E, N);
        epilogue_k<<<dim3((unsigned)((tot + 255) / 256)), blk, 0, stream>>>(A, dis, b2, B, n, N, 1);
    }
    // --- layer 3: B -> A -> scatter into B (N=64, no ReLU) ---
    {
        const int N = 64;
        const int tiles = mtiles * (N / 16);
        gemm_wmma_f32_k<64><<<dim3((tiles + 7) / 8), blk, 0, stream>>>(B, W3, A, n);
        long long tot = (long long)n * N;
        zero_k<<<dim3((unsigned)((tot + 255) / 256)), blk, 0, stream>>>(B, tot);
        long long st = (long long)E * (N / 4);
        scatter_k<<<dim3((unsigned)((st + 255) / 256)), blk, 0, stream>>>(A, src, dst, dis, B, E, N);
        epilogue_k<<<dim3((unsigned)((tot + 255) / 256)), blk, 0, stream>>>(A, dis, b3, B, n, N, 0);
    }

    rownorm_k<<<dim3((n + 7) / 8), blk, 0, stream>>>(B, xout, n);

    // --- y branch ---
    ystats_k<<<dim3(1),   dim3(256), 0, stream>>>(y, stats, 256);
    ymlp1_k <<<dim3(128), dim3(256), 0, stream>>>(y, stats, bng, bnb, Wm1, bm1, T1);
    ymlp2_k <<<dim3(256), dim3(64),  0, stream>>>(T1, Wm2, bm2, yout);
}